// F2VConv3d_54640573939775
// MI455X (gfx1250) — compile-verified
//
#include <hip/hip_runtime.h>

// Problem constants (from reference)
#define NF      200000
#define NV      100000
#define C_IN    128
#define MULT    2
#define KF      16
#define C_OUT   128
#define CM      256          // C_IN * MULT
#define BN_EPS  1e-3f

typedef float v2f __attribute__((ext_vector_type(2)));
typedef float v8f __attribute__((ext_vector_type(8)));

// D = A(16x4 f32) * B(4x16 f32) + C(16x16 f32), full fp32 — matches reference precision.
__device__ __forceinline__ v8f wmma_f32_16x16x4(v2f a, v2f b, v8f c) {
    return __builtin_amdgcn_wmma_f32_16x16x4_f32(
        /*neg_a=*/false, a, /*neg_b=*/false, b,
        /*c_mod=*/(short)0, c, /*reuse_a=*/false, /*reuse_b=*/false);
}

// ---------------------------------------------------------------------------
// K0: zero agg[NV*CM] + colsum[128] + colsumsq[128]
// ---------------------------------------------------------------------------
__global__ void zero_ws_kernel(float4* __restrict__ p, int n4) {
    for (int i = blockIdx.x * blockDim.x + threadIdx.x; i < n4;
         i += gridDim.x * blockDim.x)
        p[i] = make_float4(0.f, 0.f, 0.f, 0.f);
}

// ---------------------------------------------------------------------------
// K1: per 16-face tile: w = filt_coeff(16x16) @ sw_flat(16x256) via WMMA,
//     feat = inputs * w, scatter-add into agg via fp32 atomics.
// One wave per tile, 4 waves per block. 12500 tiles total (NF/16, exact).
// ---------------------------------------------------------------------------
__global__ void __launch_bounds__(128)
face_scatter_kernel(const float* __restrict__ inputs,    // [NF, C_IN]
                    const float* __restrict__ fc,        // [NF, KF]
                    const int*   __restrict__ face,      // [NF, 3]
                    const int*   __restrict__ vt_map,    // [NV]
                    const float* __restrict__ sw,        // [KF, CM] (K,C,M row-major)
                    float*       __restrict__ agg) {     // [NV, CM]
    __shared__ int tgts[4][48];                          // 3 target verts x 16 faces per wave

    const int lane = threadIdx.x & 31;
    const int wave = threadIdx.x >> 5;
    const int tile = blockIdx.x * 4 + wave;              // 0..12499
    const int fb   = tile * 16;
    const int m    = lane & 15;                          // A-row / B-col index
    const int hi   = lane >> 4;                          // lane half
    const int koff = hi * 2;                             // K offset within 4-chunk

    if (lane < 16) {
        #pragma unroll
        for (int j = 0; j < 3; ++j)
            tgts[wave][j * 16 + m] = vt_map[face[(fb + m) * 3 + j]];
    }
    __syncthreads();

    // A operand: filt_coeff rows, K=16 split into 4 chunks of 4
    v2f a[4];
    #pragma unroll
    for (int q = 0; q < 4; ++q) {
        const float* row = fc + (fb + m) * KF + 4 * q + koff;
        a[q].x = row[0];
        a[q].y = row[1];
    }

    for (int ct = 0; ct < 16; ++ct) {                    // 16 column tiles of CM=256
        const int cmb = ct * 16;
        v8f acc = {};
        #pragma unroll
        for (int q = 0; q < 4; ++q) {
            v2f b;
            b.x = sw[(4 * q + koff)     * CM + cmb + m];
            b.y = sw[(4 * q + koff + 1) * CM + cmb + m];
            acc = wmma_f32_16x16x4(a[q], b, acc);
        }
        // C/D layout: VGPR r -> M = r (lanes 0-15) or r+8 (lanes 16-31), N = lane&15
        #pragma unroll
        for (int r = 0; r < 8; ++r) {
            const int   M   = r + hi * 8;
            const int   f   = fb + M;
            const float val = acc[r] * inputs[f * C_IN + ((cmb + m) >> 1)];
            #pragma unroll
            for (int j = 0; j < 3; ++j) {
                const int tv = tgts[wave][j * 16 + M];
                atomicAdd(agg + tv * CM + cmb + m, val); // L2-resident (102MB < 192MB L2)
            }
        }
    }
}

// ---------------------------------------------------------------------------
// K2: out_pre = relu( (agg/denom) @ dw + bias ), write to d_out, accumulate
//     per-column sum / sumsq for BN. One wave per (16-row tile, 64-col half):
//     64 chained f32 WMMAs, acc = 4 tiles of 16x16. 12500 waves (exact).
// ---------------------------------------------------------------------------
__global__ void __launch_bounds__(128)
vert_gemm_kernel(const float* __restrict__ agg,          // [NV, CM]
                 const int*   __restrict__ nf_count,     // [NV]
                 const float* __restrict__ dw,           // [CM, C_OUT]
                 const float* __restrict__ bias,         // [C_OUT]
                 float*       __restrict__ out,          // [NV, C_OUT]
                 float*       __restrict__ colsum,       // [C_OUT]
                 float*       __restrict__ colsumsq) {   // [C_OUT]
    const int lane  = threadIdx.x & 31;
    const int wave  = threadIdx.x >> 5;
    const int wid   = blockIdx.x * 4 + wave;             // 0..12499
    const int vtile = wid >> 1;                          // 0..6249
    const int half  = wid & 1;                           // which 64 output columns
    const int vb    = vtile * 16;
    const int n     = lane & 15;
    const int hi    = lane >> 4;
    const int koff  = hi * 2;

    // Fold per-row mean-divide (1/max(nf_count,1)) into the A operand.
    const int   arow = vb + n;                           // A-matrix row this lane feeds
    const int   cnt  = nf_count[arow];
    const float rd   = 1.0f / (float)(cnt > 0 ? cnt : 1);

    v8f acc[4];
    v8f vzero = {};
    #pragma unroll
    for (int t = 0; t < 4; ++t) acc[t] = vzero;

    for (int q = 0; q < 64; ++q) {                       // K = 256 = 64 chunks of 4
        v2f a;
        const float* ar = agg + arow * CM + 4 * q + koff;
        a.x = ar[0] * rd;
        a.y = ar[1] * rd;
        #pragma unroll
        for (int t = 0; t < 4; ++t) {
            const int cb = (half * 4 + t) * 16;
            v2f b;
            b.x = dw[(4 * q + koff)     * C_OUT + cb + n];
            b.y = dw[(4 * q + koff + 1) * C_OUT + cb + n];
            acc[t] = wmma_f32_16x16x4(a, b, acc[t]);
        }
    }

    // Epilogue: bias + ReLU, store, and BN statistics.
    #pragma unroll
    for (int t = 0; t < 4; ++t) {
        const int   cb = (half * 4 + t) * 16;
        const float bv = bias[cb + n];
        float s = 0.f, s2 = 0.f;
        #pragma unroll
        for (int r = 0; r < 8; ++r) {
            const int M = r + hi * 8;
            float v = acc[t][r] + bv;
            v = v > 0.f ? v : 0.f;
            out[(vb + M) * C_OUT + cb + n] = v;
            s  += v;
            s2 += v * v;
        }
        // lanes n and n+16 hold the same column: combine, then one atomic each
        s  += __shfl_xor(s, 16);
        s2 += __shfl_xor(s2, 16);
        if (hi == 0) {
            atomicAdd(colsum   + cb + n, s);
            atomicAdd(colsumsq + cb + n, s2);
        }
    }
}

// ---------------------------------------------------------------------------
// K3: BatchNorm finalize in-place on d_out (biased variance, eps=1e-3)
// ---------------------------------------------------------------------------
__global__ void bn_finalize_kernel(float* __restrict__ out,
                                   const float* __restrict__ colsum,
                                   const float* __restrict__ colsumsq,
                                   const float* __restrict__ gamma,
                                   const float* __restrict__ beta) {
    const float invN = 1.0f / (float)NV;
    const int total = NV * C_OUT;
    for (int idx = blockIdx.x * blockDim.x + threadIdx.x; idx < total;
         idx += gridDim.x * blockDim.x) {
        const int   c     = idx & (C_OUT - 1);
        const float mean  = colsum[c] * invN;
        const float var   = colsumsq[c] * invN - mean * mean;
        const float scale = gamma[c] * rsqrtf(var + BN_EPS);
        out[idx] = (out[idx] - mean) * scale + beta[c];
    }
}

// ---------------------------------------------------------------------------
extern "C" void kernel_launch(void* const* d_in, const int* in_sizes, int n_in,
                              void* d_out, int out_size, void* d_ws, size_t ws_size,
                              hipStream_t stream) {
    (void)in_sizes; (void)n_in; (void)out_size; (void)ws_size;
    const float* inputs   = (const float*)d_in[0];
    const float* fc       = (const float*)d_in[1];
    const int*   face     = (const int*)  d_in[2];
    const int*   nf_count = (const int*)  d_in[3];
    const int*   vt_map   = (const int*)  d_in[4];
    const float* sw       = (const float*)d_in[5];
    const float* dw       = (const float*)d_in[6];
    const float* bias     = (const float*)d_in[7];
    const float* gamma    = (const float*)d_in[8];
    const float* beta     = (const float*)d_in[9];
    float* out = (float*)d_out;

    float* agg      = (float*)d_ws;                 // NV*CM floats (102.4 MB)
    float* colsum   = agg + (size_t)NV * CM;        // 128 floats
    float* colsumsq = colsum + C_OUT;               // 128 floats

    const int n4 = (NV * CM + 2 * C_OUT) / 4;       // 6,400,064 float4s
    zero_ws_kernel<<<2048, 256, 0, stream>>>((float4*)d_ws, n4);

    face_scatter_kernel<<<NF / 16 / 4, 128, 0, stream>>>(inputs, fc, face, vt_map, sw, agg);

    vert_gemm_kernel<<<(NV / 16) * 2 / 4, 128, 0, stream>>>(agg, nf_count, dw, bias,
                                                            out, colsum, colsumsq);

    bn_finalize_kernel<<<4096, 256, 0, stream>>>(out, colsum, colsumsq, gamma, beta);
}